// SigmoidWarpageLoss_20461224198290
// MI455X (gfx1250) — compile-verified
//
#include <hip/hip_runtime.h>
#include <hip/hip_bf16.h>

// SigmoidWarpageLoss on MI455X (gfx1250, wave32).
// Streaming elementwise + scalar reduction:
//   160 MB logits read once -> ~7us HBM floor @ 23.3 TB/s.
// Pipeline (all on `stream`):
//   (0) memsetAsync max slot
//   (1) max_label_kernel : global max of (cls>=1 ? iou : 0) via shuffle + atomicMax(bits)
//   (2) main_kernel      : b128 NT loads, stable softplus/sigmoid, per-block partial
//                          (wave reduce via V_WMMA_F32_16X16X4_F32 with B = ones)
//   (3) final_kernel     : deterministic single-block tree reduce -> d_out[0]

typedef float v2f __attribute__((ext_vector_type(2)));
typedef float v4f __attribute__((ext_vector_type(4)));
typedef float v8f __attribute__((ext_vector_type(8)));

#define NROWS 500000
#define NCOLS 80
#define NQUADS ((NROWS * NCOLS) / 4)   // 10,000,000 float4 tiles, no tail
#define QPR (NCOLS / 4)                // 20 float4 per row
#define MAIN_BLOCKS 2048

// ---------------- Kernel 1: global max of label values -----------------------
__global__ void max_label_kernel(const float* __restrict__ targets,
                                 unsigned* __restrict__ maxBits, int n) {
  int tid = blockIdx.x * blockDim.x + threadIdx.x;
  int stride = gridDim.x * blockDim.x;
  float m = 0.0f;                       // label matrix is >= 0; 0 is the identity
  for (int i = tid; i < n; i += stride) {
    float c = targets[2 * i];           // class as float, 0..80; fg iff c >= 1
    float u = targets[2 * i + 1];       // iou in [0,1)
    if (c >= 1.0f) m = fmaxf(m, u);
  }
  // wave32 max tree
  #pragma unroll
  for (int off = 16; off > 0; off >>= 1)
    m = fmaxf(m, __shfl_xor(m, off, 32));
  if ((threadIdx.x & 31) == 0)
    atomicMax(maxBits, __float_as_uint(m));   // bit order == float order for >= 0
}

// ---------------- Kernel 2: elementwise loss + block partial sums ------------
__global__ void main_kernel(const v4f* __restrict__ logits4,
                            const v2f* __restrict__ targets2,
                            const unsigned* __restrict__ maxBits,
                            float* __restrict__ partials) {
  const float maxv = __uint_as_float(*maxBits);
  const float invMax = 1.0f / maxv;     // one IEEE divide per thread

  float acc = 0.0f;
  int tid = blockIdx.x * blockDim.x + threadIdx.x;
  int stride = gridDim.x * blockDim.x;

  for (int v = tid; v < NQUADS; v += stride) {
    int row = v / QPR;
    int j0 = (v - row * QPR) * 4;
    v4f x4 = __builtin_nontemporal_load(&logits4[v]);   // streamed once: NT
    v2f tg = targets2[row];                             // reused/broadcast: cached
    int cls = (int)tg[0] - 1;                           // -1..79; -1 never matches
    float scaled = tg[1] * invMax;

    #pragma unroll
    for (int k = 0; k < 4; ++k) {
      float x = x4[k];
      float ax = fabsf(x);
      float e = __expf(-ax);                 // v_exp_f32
      float den = 1.0f + e;
      float r = __builtin_amdgcn_rcpf(den);  // v_rcp_f32
      float p = (x >= 0.0f) ? r : e * r;     // sigmoid(x)
      float t = __logf(den);                 // softplus(-|x|) = log1p(e)
      float bce0 = fmaxf(x, 0.0f) + t;       // softplus(x)
      float bce1 = fmaxf(-x, 0.0f) + t;      // softplus(-x)
      float lbl = (cls == j0 + k) ? scaled : 0.0f;
      float tneg = (bce0 * (1.0f - lbl) + (lbl - p)) * 0.75f;
      float tpos = (bce1 * lbl + (p - lbl)) * 0.25f;
      acc += (p <= lbl) ? tpos : tneg;
    }
  }

  // ---- Wave32 sum via WMMA f32 16x16x4: D = A(16x4) x ones(4x16) -----------
  // A vgpr0: lanes0-15 -> A[m][0]=acc[m]; lanes16-31 -> A[m][2]=acc[m+16].
  // A vgpr1: K=1/K=3 = 0.  B = all ones  =>  D[m][n] = acc[m] + acc[m+16].
  // EXEC is all ones here (no early exits before this point).
  v2f a; a[0] = acc;  a[1] = 0.0f;
  v2f b; b[0] = 1.0f; b[1] = 1.0f;
  v8f c = {0.f, 0.f, 0.f, 0.f, 0.f, 0.f, 0.f, 0.f};
  v8f d = __builtin_amdgcn_wmma_f32_16x16x4_f32(
      /*neg_a=*/false, a, /*neg_b=*/false, b,
      /*c_mod=*/(short)0, c, /*reuse_a=*/false, /*reuse_b=*/false);
  // lane n in 0..15 holds D rows 0..7 (sum of acc[0..7]+acc[16..23]);
  // lane 16+n holds rows 8..15 (sum of acc[8..15]+acc[24..31]).
  float s = d[0] + d[1] + d[2] + d[3] + d[4] + d[5] + d[6] + d[7];
  s += __shfl_xor(s, 16, 32);            // full wave total in every lane

  __shared__ float wsum[8];
  int lane = threadIdx.x & 31;
  int wave = threadIdx.x >> 5;
  if (lane == 0) wsum[wave] = s;
  __syncthreads();
  if (threadIdx.x == 0) {
    float bs = 0.0f;
    #pragma unroll
    for (int w = 0; w < 8; ++w) bs += wsum[w];   // fixed order: deterministic
    partials[blockIdx.x] = bs;
  }
}

// ---------------- Kernel 3: deterministic final reduction --------------------
__global__ void final_kernel(const float* __restrict__ partials, int nb,
                             float* __restrict__ out) {
  __shared__ float sm[256];
  float acc = 0.0f;
  for (int i = threadIdx.x; i < nb; i += 256) acc += partials[i];
  sm[threadIdx.x] = acc;
  __syncthreads();
  #pragma unroll
  for (int off = 128; off > 0; off >>= 1) {
    if (threadIdx.x < off) sm[threadIdx.x] += sm[threadIdx.x + off];
    __syncthreads();
  }
  if (threadIdx.x == 0) out[0] = sm[0];
}

extern "C" void kernel_launch(void* const* d_in, const int* in_sizes, int n_in,
                              void* d_out, int out_size, void* d_ws, size_t ws_size,
                              hipStream_t stream) {
  (void)in_sizes; (void)n_in; (void)out_size; (void)ws_size;
  const float* logits  = (const float*)d_in[0];   // (500000, 80) f32
  const float* targets = (const float*)d_in[1];   // (500000, 2)  f32
  float* out = (float*)d_out;                     // 1 x f32

  float* ws = (float*)d_ws;
  unsigned* maxBits = (unsigned*)ws;              // ws[0]: max label bits
  float* partials = ws + 64;                      // ws[64..64+MAIN_BLOCKS)

  (void)hipMemsetAsync(d_ws, 0, sizeof(unsigned), stream);  // max identity = +0.0f
  max_label_kernel<<<1024, 256, 0, stream>>>(targets, maxBits, NROWS);
  main_kernel<<<MAIN_BLOCKS, 256, 0, stream>>>(
      (const v4f*)logits, (const v2f*)targets, maxBits, partials);
  final_kernel<<<1, 256, 0, stream>>>(partials, MAIN_BLOCKS, out);
}